// NeuralNet_19516331393457
// MI455X (gfx1250) — compile-verified
//
#include <hip/hip_runtime.h>

#define NIN   128   // INPUT_SIZE
#define NH1   128
#define NH2   64
#define NVOC  48

typedef __attribute__((ext_vector_type(2))) float v2f;
typedef __attribute__((ext_vector_type(8))) float v8f;

// ---------------------------------------------------------------------------
// Kernel 0: zero the argmax scratch (u64 packed (key<<32)|~index)
// ---------------------------------------------------------------------------
__global__ void nn_init_kernel(unsigned long long* wsm) {
    if (threadIdx.x == 0 && blockIdx.x == 0) wsm[0] = 0ull;
}

// ---------------------------------------------------------------------------
// Kernel 1: main MLP + F + argmax reduction.
// 128 threads = 4 waves; each wave handles a 16-sample tile with fp32 WMMA.
// ---------------------------------------------------------------------------
__global__ __launch_bounds__(128) void nn_mlp_argmax_kernel(
    const int*   __restrict__ data,
    const float* __restrict__ embed,
    const float* __restrict__ W1, const float* __restrict__ b1,
    const float* __restrict__ W2, const float* __restrict__ b2,
    const float* __restrict__ W3, const float* __restrict__ b3,
    unsigned long long* __restrict__ wsm,
    int nsamp)
{
    __shared__ float s_embed[NVOC];
    __shared__ float s_h1[4][16][132];   // padded stride: 132 % 64 == 4 -> conflict-free
    __shared__ float s_h2[4][16][68];    // padded stride: 68  % 64 == 4
    __shared__ unsigned long long s_red[4];

    const int tid = threadIdx.x;
    if (tid < NVOC) s_embed[tid] = embed[tid];
    __syncthreads();

    const int wave = tid >> 5;
    const int lane = tid & 31;
    const int half = lane >> 4;    // 0: K 0..1 / rows 0..7 of C ; 1: K 2..3 / rows 8..15
    const int n    = lane & 15;    // A-row m, or B/C column within 16-wide tile

    const int srow  = (blockIdx.x * 4 + wave) * 16 + n;          // true sample index
    const int srowC = (srow < nsamp) ? srow : (nsamp - 1);       // clamped for loads

    // ---- gather x A-fragments: frag[kk] = { x[m][4kk+2h], x[m][4kk+2h+1] } ----
    v2f frag[32];
    {
        const int* drow = data + (size_t)srowC * NIN + 2 * half;
        #pragma unroll
        for (int kk = 0; kk < 32; ++kk) {
            int i0 = drow[4 * kk];
            int i1 = drow[4 * kk + 1];
            frag[kk].x = s_embed[i0];
            frag[kk].y = s_embed[i1];
        }
    }

    // ---- Layer 1: h1 = relu(x @ W1 + b1), 16x128 tile ----
    float (*h1w)[132] = s_h1[wave];
    for (int nt = 0; nt < 8; ++nt) {
        v8f c = {};
        const float* Wb = W1 + (2 * half) * NH1 + nt * 16 + n;
        #pragma unroll
        for (int kk = 0; kk < 32; ++kk) {
            v2f bfr;
            bfr.x = Wb[(4 * kk    ) * NH1];
            bfr.y = Wb[(4 * kk + 1) * NH1];
            c = __builtin_amdgcn_wmma_f32_16x16x4_f32(
                    false, frag[kk], false, bfr, (short)0, c, false, false);
        }
        const float bias = b1[nt * 16 + n];
        #pragma unroll
        for (int v = 0; v < 8; ++v) {
            float h = c[v] + bias;
            h1w[v + 8 * half][nt * 16 + n] = (h > 0.0f) ? h : 0.0f;
        }
    }
    __syncthreads();

    // ---- reload h1 as A-fragments ----
    #pragma unroll
    for (int kk = 0; kk < 32; ++kk) {
        frag[kk].x = h1w[n][4 * kk + 2 * half];
        frag[kk].y = h1w[n][4 * kk + 2 * half + 1];
    }

    // ---- Layer 2: h2 = tanh(h1 @ W2 + b2), 16x64 tile ----
    float (*h2w)[68] = s_h2[wave];
    for (int nt = 0; nt < 4; ++nt) {
        v8f c = {};
        const float* Wb = W2 + (2 * half) * NH2 + nt * 16 + n;
        #pragma unroll
        for (int kk = 0; kk < 32; ++kk) {
            v2f bfr;
            bfr.x = Wb[(4 * kk    ) * NH2];
            bfr.y = Wb[(4 * kk + 1) * NH2];
            c = __builtin_amdgcn_wmma_f32_16x16x4_f32(
                    false, frag[kk], false, bfr, (short)0, c, false, false);
        }
        const float bias = b2[nt * 16 + n];
        #pragma unroll
        for (int v = 0; v < 8; ++v) {
            h2w[v + 8 * half][nt * 16 + n] = tanhf(c[v] + bias);
        }
    }
    __syncthreads();

    // ---- Layer 3: out[m][c] with c = half, m = n (tiny, scalar) ----
    float acc = b3[half];
    {
        const float* h2row = h2w[n];
        #pragma unroll
        for (int k = 0; k < NH2; ++k)
            acc += h2row[k] * W3[2 * k + half];
    }
    // lane m gets out0 from lane m (half 0) and out1 from lane m+16 (half 1)
    const float x = __shfl(acc, n, 32);
    const float y = __shfl(acc, n + 16, 32);

    // ---- pred = F(x, y) ----
    const float x2 = x * x, y2 = y * y;
    const float t1 = 3.0f * (1.0f - x) * (1.0f - x) * __expf(-x2 - (y + 1.0f) * (y + 1.0f));
    const float t2 = -10.0f * (0.2f * x - x * x * x - y * y * y * y * y) * __expf(-x2 - y2);
    // 1/(3**exp(-(x+1)^2 - y^2)) = exp(-ln(3) * exp(-(x+1)^2 - y^2))
    const float t3 = -__expf(-1.09861228866810969f * __expf(-(x + 1.0f) * (x + 1.0f) - y2));
    const float pred = t1 + t2 + t3;

    // ---- pack (monotone float key, ~index) so max == (argmax, first occurrence) ----
    unsigned int kb = __float_as_uint(pred);
    kb = (kb & 0x80000000u) ? ~kb : (kb | 0x80000000u);
    unsigned long long packed =
        ((unsigned long long)kb << 32) | (unsigned int)(~(unsigned int)srow);
    if (srow >= nsamp) packed = 0ull;   // tail rows excluded (both halves of a row agree)

    // wave reduction
    #pragma unroll
    for (int off = 16; off >= 1; off >>= 1) {
        unsigned long long other = __shfl_xor(packed, off, 32);
        if (other > packed) packed = other;
    }
    if (lane == 0) s_red[wave] = packed;
    __syncthreads();
    if (wave == 0) {
        unsigned long long p = (lane < 4) ? s_red[lane] : 0ull;
        #pragma unroll
        for (int off = 2; off >= 1; off >>= 1) {
            unsigned long long other = __shfl_xor(p, off, 32);
            if (other > p) p = other;
        }
        if (lane == 0) atomicMax(wsm, p);
    }
}

// ---------------------------------------------------------------------------
// Kernel 2: recompute out[2] for the winning sample exactly, write d_out.
// ---------------------------------------------------------------------------
__global__ __launch_bounds__(128) void nn_finalize_kernel(
    const int*   __restrict__ data,
    const float* __restrict__ embed,
    const float* __restrict__ W1, const float* __restrict__ b1,
    const float* __restrict__ W2, const float* __restrict__ b2,
    const float* __restrict__ W3, const float* __restrict__ b3,
    const unsigned long long* __restrict__ wsm,
    float* __restrict__ out)
{
    __shared__ float sx[NIN];
    __shared__ float sh1[NH1];
    __shared__ float sh2[NH2];

    const unsigned long long packed = wsm[0];
    const unsigned int idx = ~(unsigned int)(packed & 0xFFFFFFFFull);
    const int t = threadIdx.x;   // 128 threads

    sx[t] = embed[data[(size_t)idx * NIN + t]];
    __syncthreads();

    float a = b1[t];
    for (int k = 0; k < NIN; ++k) a += sx[k] * W1[k * NH1 + t];
    sh1[t] = (a > 0.0f) ? a : 0.0f;
    __syncthreads();

    if (t < NH2) {
        float a2 = b2[t];
        for (int k = 0; k < NH1; ++k) a2 += sh1[k] * W2[k * NH2 + t];
        sh2[t] = tanhf(a2);
    }
    __syncthreads();

    if (t < 2) {
        float a3 = b3[t];
        for (int k = 0; k < NH2; ++k) a3 += sh2[k] * W3[k * 2 + t];
        out[t] = a3;
    }
}

// ---------------------------------------------------------------------------
extern "C" void kernel_launch(void* const* d_in, const int* in_sizes, int n_in,
                              void* d_out, int out_size, void* d_ws, size_t ws_size,
                              hipStream_t stream) {
    const int*   data  = (const int*)  d_in[0];
    const float* embed = (const float*)d_in[1];
    const float* W1    = (const float*)d_in[2];
    const float* b1    = (const float*)d_in[3];
    const float* W2    = (const float*)d_in[4];
    const float* b2    = (const float*)d_in[5];
    const float* W3    = (const float*)d_in[6];
    const float* b3    = (const float*)d_in[7];

    const int nsamp = in_sizes[0] / NIN;   // 500000
    unsigned long long* wsm = (unsigned long long*)d_ws;

    nn_init_kernel<<<1, 32, 0, stream>>>(wsm);

    const int blocks = (nsamp + 63) / 64;  // 64 samples per 128-thread block (4 waves x 16)
    nn_mlp_argmax_kernel<<<blocks, 128, 0, stream>>>(
        data, embed, W1, b1, W2, b2, W3, b3, wsm, nsamp);

    nn_finalize_kernel<<<1, 128, 0, stream>>>(
        data, embed, W1, b1, W2, b2, W3, b3, wsm, (float*)d_out);
}